// WaveletDecomposer_65017214927375
// MI455X (gfx1250) — compile-verified
//
#include <hip/hip_runtime.h>

typedef float v2f __attribute__((ext_vector_type(2)));
typedef float v8f __attribute__((ext_vector_type(8)));

__device__ __constant__ float g_dec_lo[8] = {
    -0.010597401784997278f, 0.032883011666982945f, 0.030841381835986965f,
    -0.18703481171888114f, -0.02798376941698385f, 0.6308807679295904f,
    0.7148465705525415f, 0.23037781330885523f};
__device__ __constant__ float g_dec_hi[8] = {
    -0.23037781330885523f, 0.7148465705525415f, -0.6308807679295904f,
    -0.02798376941698385f, 0.18703481171888114f, 0.030841381835986965f,
    -0.032883011666982945f, -0.010597401784997278f};

// Level lengths: 4096 -> 2051 -> 1029 -> 518 -> 262 -> 134
// Output layout (floats): approx@0(134), d5@548864(134), d4@1097728(262),
//                         d3@2170880(518), d2@4292608(1029), d1@8507392(2051)

__global__ __launch_bounds__(256)
void WaveletDecomposer_kernel(const float* __restrict__ x, float* __restrict__ out) {
    constexpr int NLVL = 5;
    constexpr int T0 = 4096;
    const int lens[NLVL + 1] = {4096, 2051, 1029, 518, 262, 134};
    const long long doff[NLVL] = {8507392LL, 4292608LL, 2170880LL, 1097728LL, 548864LL};

    // ext buffer layout: data x[t] lives at buf[8+t]; left halo buf[2..7],
    // right halo buf[8+L .. 8+L+6]. ext[j] = buf[2+j].
    __shared__ __align__(16) float bufA[4112];   // 4096 + 16
    __shared__ __align__(16) float bufB[2068];   // 2051 + 16 (rounded)

    const int row  = blockIdx.x;
    const int tid  = threadIdx.x;
    const int lane = tid & 31;
    const int wave = tid >> 5;
    const int m    = lane & 15;   // M (filter row) for A; N (position) for B
    const int half = lane >> 4;   // K-half selector

    // --- A operand: reversed filter matrix, rows 0/1 used, rest zero -------
    // Lane m holds A[m, 2*half + {0,1}] for WMMA#1 (taps 0..3) and
    // A[m, 4 + 2*half + {0,1}] for WMMA#2 (taps 4..7); frev[k] = filt[7-k].
    v2f A1; A1.x = 0.f; A1.y = 0.f;
    v2f A2; A2.x = 0.f; A2.y = 0.f;
    if (m < 2) {
        const float* f = (m == 0) ? g_dec_lo : g_dec_hi;
        const int k0 = half * 2;
        A1.x = f[7 - k0];
        A1.y = f[6 - k0];
        A2.x = f[3 - k0];
        A2.y = f[2 - k0];
    }

    // --- stage the full row into LDS (single HBM pass) ---------------------
    {
        const float4* xr = (const float4*)(x + (size_t)row * T0);
        float4* s4 = (float4*)(bufA + 8);              // 32B aligned
        for (int i = tid; i < T0 / 4; i += 256) s4[i] = xr[i];
    }
    __syncthreads();

    // halo fill: left = x[5..0] reversed, right = x[L-1..L-7] reversed
    if (tid < 6)        bufA[2 + tid] = bufA[8 + 5 - tid];
    else if (tid < 13)  { int t = tid - 6; bufA[8 + T0 + t] = bufA[8 + T0 - 1 - t]; }
    __syncthreads();

    float* bufs[2] = {bufA, bufB};

    for (int lvl = 0; lvl < NLVL; ++lvl) {
        float* __restrict__ inb  = bufs[lvl & 1];
        float* __restrict__ outb = bufs[(lvl + 1) & 1];
        const int L    = lens[lvl];
        const int olen = lens[lvl + 1];
        const unsigned cap = (unsigned)(L + 14);       // last valid buf index
        float* __restrict__ gD = out + doff[lvl] + (size_t)row * olen;
        const int ntiles = (olen + 15) >> 4;

        for (int t = wave; t < ntiles; t += 8) {       // wave-uniform bounds
            const int base = t << 4;
            const int pos  = base + m;
            // B operand: lane holds ext[2*pos + 2*half + {0,1}] (WMMA#1)
            //            and ext[2*pos + 4 + 2*half + {0,1}] (WMMA#2)
            unsigned jb = (unsigned)(2 + 2 * pos + 2 * half);
            unsigned j0 = jb      < cap ? jb      : cap;
            unsigned j1 = jb + 1u < cap ? jb + 1u : cap;
            unsigned j2 = jb + 4u < cap ? jb + 4u : cap;
            unsigned j3 = jb + 5u < cap ? jb + 5u : cap;
            v2f B1; B1.x = inb[j0]; B1.y = inb[j1];
            v2f B2; B2.x = inb[j2]; B2.y = inb[j3];

            v8f c = {};
            c = __builtin_amdgcn_wmma_f32_16x16x4_f32(false, A1, false, B1,
                                                      (short)0, c, false, false);
            c = __builtin_amdgcn_wmma_f32_16x16x4_f32(false, A2, false, B2,
                                                      (short)0, c, false, false);
            // D: VGPR0 lanes 0-15 = cA[base+lane], VGPR1 lanes 0-15 = cD[...]
            if (half == 0 && pos < olen) {
                outb[8 + pos] = c[0];                  // cA -> next level input
                gD[pos] = c[1];                        // detail band
                if (lvl == NLVL - 1)
                    out[(size_t)row * 134 + pos] = c[0];  // final approx
            }
        }
        __syncthreads();

        if (lvl < NLVL - 1) {                          // halo for next level
            if (tid < 6)       outb[2 + tid] = outb[8 + 5 - tid];
            else if (tid < 13) { int t2 = tid - 6; outb[8 + olen + t2] = outb[8 + olen - 1 - t2]; }
            __syncthreads();
        }
    }
}

extern "C" void kernel_launch(void* const* d_in, const int* in_sizes, int n_in,
                              void* d_out, int out_size, void* d_ws, size_t ws_size,
                              hipStream_t stream) {
    (void)in_sizes; (void)n_in; (void)d_ws; (void)ws_size; (void)out_size;
    const float* x = (const float*)d_in[0];
    float* out = (float*)d_out;
    dim3 grid(64 * 64);    // one block per (b, c) row
    dim3 block(256);       // 8 wave32s
    WaveletDecomposer_kernel<<<grid, block, 0, stream>>>(x, out);
}